// GCNDeepSet_72954314490133
// MI455X (gfx1250) — compile-verified
//
#include <hip/hip_runtime.h>
#include <hip/hip_bf16.h>

typedef __attribute__((ext_vector_type(2))) float v2f;
typedef __attribute__((ext_vector_type(8))) float v8f;

static __device__ __forceinline__ void atomAddF(float* p, float v) {
    (void)__hip_atomic_fetch_add(p, v, __ATOMIC_RELAXED, __HIP_MEMORY_SCOPE_AGENT);
}

// ---------------------------------------------------------------------------
// zero a float buffer
__global__ void zero_k(float* __restrict__ p, int n) {
    int i = blockIdx.x * blockDim.x + threadIdx.x;
    if (i < n) p[i] = 0.0f;
}

// ---------------------------------------------------------------------------
// Collapse the affine DeepSet branches:
//   M = phi_W @ rho_W  (2x6),  c = phi_b @ rho_W  (6)
// consts layout: [0..11] M_A (row-major 2x6), [16..21] c_A,
//                [32..43] M_T,               [48..53] c_T
__global__ void precompute_consts(const float* __restrict__ phiA_W,
                                  const float* __restrict__ phiA_b,
                                  const float* __restrict__ rhoA_W,
                                  const float* __restrict__ phiT_W,
                                  const float* __restrict__ phiT_b,
                                  const float* __restrict__ rhoT_W,
                                  float* __restrict__ consts, int H) {
    int t = threadIdx.x;  // 64 threads
    if (t < 12) {
        int j = t / 6, o = t % 6;
        float s = 0.f;
        for (int h = 0; h < H; ++h) s += phiA_W[j * H + h] * rhoA_W[h * 6 + o];
        consts[t] = s;
    } else if (t >= 16 && t < 22) {
        int o = t - 16;
        float s = 0.f;
        for (int h = 0; h < H; ++h) s += phiA_b[h] * rhoA_W[h * 6 + o];
        consts[t] = s;
    } else if (t >= 32 && t < 44) {
        int tt = t - 32;
        int j = tt / 6, o = tt % 6;
        float s = 0.f;
        for (int h = 0; h < H; ++h) s += phiT_W[j * H + h] * rhoT_W[h * 6 + o];
        consts[t] = s;
    } else if (t >= 48 && t < 54) {
        int o = t - 48;
        float s = 0.f;
        for (int h = 0; h < H; ++h) s += phiT_b[h] * rhoT_W[h * 6 + o];
        consts[t] = s;
    }
}

// ---------------------------------------------------------------------------
// Segment-sum of 2D source positions + degree count per destination node.
// Used for both agent-agent edges and target-agent edges.
__global__ void edge_accum(const int* __restrict__ esrc, const int* __restrict__ edst,
                           const float* __restrict__ pos2,
                           float* __restrict__ S, float* __restrict__ deg, int E) {
    int e = blockIdx.x * blockDim.x + threadIdx.x;
    if (e >= E) return;
    int i0 = esrc[e];
    int i1 = edst[e];
    atomAddF(&S[i1 * 2 + 0], pos2[i0 * 2 + 0]);
    atomAddF(&S[i1 * 2 + 1], pos2[i0 * 2 + 1]);
    atomAddF(&deg[i1], 1.0f);
}

// ---------------------------------------------------------------------------
// Per-node: reconstruct a (6), t (6) from collapsed DeepSets, then
// x = [own(4), a(6), t(6)] @ rin_W(16x32) + rin_b
__global__ void node_features(const float* __restrict__ own_obs,
                              const float* __restrict__ agent_pos,
                              const float* __restrict__ S_A, const float* __restrict__ degA,
                              const float* __restrict__ S_T, const float* __restrict__ degT,
                              const float* __restrict__ consts,
                              const float* __restrict__ rhoA_b,
                              const float* __restrict__ rhoT_b,
                              const float* __restrict__ rin_W,
                              const float* __restrict__ rin_b,
                              float* __restrict__ xbuf, int N) {
    __shared__ float sW[16 * 32];
    __shared__ float sB[32];
    __shared__ float sC[64];
    for (int t = threadIdx.x; t < 16 * 32; t += blockDim.x) sW[t] = rin_W[t];
    if (threadIdx.x < 32) sB[threadIdx.x] = rin_b[threadIdx.x];
    if (threadIdx.x < 64) sC[threadIdx.x] = consts[threadIdx.x];
    __syncthreads();

    int i = blockIdx.x * blockDim.x + threadIdx.x;
    if (i >= N) return;

    float px = agent_pos[i * 2 + 0];
    float py = agent_pos[i * 2 + 1];
    float dA = degA[i];
    float fa0 = S_A[i * 2 + 0] - dA * px;
    float fa1 = S_A[i * 2 + 1] - dA * py;
    float dT = degT[i];
    float ft0 = S_T[i * 2 + 0] - dT * px;
    float ft1 = S_T[i * 2 + 1] - dT * py;

    float in[16];
    in[0] = own_obs[i * 4 + 0];
    in[1] = own_obs[i * 4 + 1];
    in[2] = own_obs[i * 4 + 2];
    in[3] = own_obs[i * 4 + 3];
#pragma unroll
    for (int o = 0; o < 6; ++o)
        in[4 + o] = fa0 * sC[o] + fa1 * sC[6 + o] + dA * sC[16 + o] + rhoA_b[o];
#pragma unroll
    for (int o = 0; o < 6; ++o)
        in[10 + o] = ft0 * sC[32 + o] + ft1 * sC[38 + o] + dT * sC[48 + o] + rhoT_b[o];

#pragma unroll
    for (int c = 0; c < 32; ++c) {
        float acc = sB[c];
#pragma unroll
        for (int j = 0; j < 16; ++j) acc += in[j] * sW[j * 32 + c];
        xbuf[i * 32 + c] = acc;
    }
}

// ---------------------------------------------------------------------------
// One graph-propagation tap: snext[g1] += w[e] * sprev[g0], 32 channels.
// One lane per channel -> rows are 128B coalesced; scatter via hw f32 atomics.
__global__ void propagate(const float* __restrict__ sprev, float* __restrict__ snext,
                          const int* __restrict__ g0, const int* __restrict__ g1,
                          const float* __restrict__ w, int E) {
    int t = blockIdx.x * blockDim.x + threadIdx.x;
    int e = t >> 5;
    if (e >= E) return;
    int lane = t & 31;
    int a = g0[e];
    int b = g1[e];
    float we = w[e];
    float v = we * sprev[a * 32 + lane];
    atomAddF(&snext[b * 32 + lane], v);
}

// ---------------------------------------------------------------------------
// Z[Npad x 32] (+)= S[Npad x 32] @ W[32 x 32]  using V_WMMA_F32_16X16X4_F32.
// One wave computes a 16-row stripe (two 16x16 col tiles). Block = 128 thr =
// 4 waves = 64 rows; Npad is a multiple of 64 so EXEC is always all-ones.
// W is staged in LDS *transposed* with stride 34 so each lane's B K-pair is
// one aligned ds_load_b64; all 16 B fragments are hoisted out of the K loop.
// If initZ: Z = S@W + bias (bias broadcast over rows), else accumulate.
__global__ void __launch_bounds__(128) gemm32_wmma(const float* __restrict__ S,
                                                   const float* __restrict__ W,
                                                   const float* __restrict__ bias,
                                                   float* __restrict__ Z, int initZ) {
    __shared__ __align__(16) float Wt[32 * 34];  // Wt[n*34 + k] = W[k*32 + n]
    __shared__ float Bl[32];
    for (int t = threadIdx.x; t < 32 * 32; t += blockDim.x) {
        int k = t >> 5, n = t & 31;
        Wt[n * 34 + k] = W[t];
    }
    if (threadIdx.x < 32) Bl[threadIdx.x] = initZ ? bias[threadIdx.x] : 0.0f;
    __syncthreads();

    const int lane = threadIdx.x & 31;
    const int wid  = threadIdx.x >> 5;
    const int r0   = (blockIdx.x * 4 + wid) * 16;
    const int n0   = lane & 15;            // col within tile / A row within tile
    const int kh   = (lane >> 4) * 2;      // K sub-offset for A/B register layout
    const int mrow = r0 + (lane >> 4) * 8; // C/D row base for this lane half

    // Hoist all B fragments (wave-invariant): 8 K-chunks x 2 col tiles.
    v2f bf0[8], bf1[8];
#pragma unroll
    for (int c = 0; c < 8; ++c) {
        int k = c * 4 + kh;  // even -> 8B-aligned LDS address (stride 34 even)
        bf0[c] = *(const v2f*)&Wt[n0 * 34 + k];
        bf1[c] = *(const v2f*)&Wt[(n0 + 16) * 34 + k];
    }

    v8f acc0, acc1;
    if (initZ) {
        float b0 = Bl[n0], b1 = Bl[n0 + 16];
#pragma unroll
        for (int v = 0; v < 8; ++v) { acc0[v] = b0; acc1[v] = b1; }
    } else {
#pragma unroll
        for (int v = 0; v < 8; ++v) {
            acc0[v] = Z[(mrow + v) * 32 + n0];
            acc1[v] = Z[(mrow + v) * 32 + n0 + 16];
        }
    }

    const float* srow = S + (size_t)(r0 + n0) * 32 + kh;  // even offset -> 8B aligned
#pragma unroll
    for (int c = 0; c < 8; ++c) {
        v2f a = *(const v2f*)(srow + c * 4);
        acc0 = __builtin_amdgcn_wmma_f32_16x16x4_f32(false, a, false, bf0[c], (short)0,
                                                     acc0, false, false);
        acc1 = __builtin_amdgcn_wmma_f32_16x16x4_f32(false, a, false, bf1[c], (short)0,
                                                     acc1, false, false);
    }

#pragma unroll
    for (int v = 0; v < 8; ++v) {
        Z[(mrow + v) * 32 + n0]      = acc0[v];
        Z[(mrow + v) * 32 + n0 + 16] = acc1[v];
    }
}

// ---------------------------------------------------------------------------
__global__ void lrelu_k(const float* __restrict__ z, float* __restrict__ x, int n) {
    int i = blockIdx.x * blockDim.x + threadIdx.x;
    if (i >= n) return;
    float v = z[i];
    x[i] = v > 0.0f ? v : 0.01f * v;
}

// out = x @ rout_W(32x2) + rout_b
__global__ void out_k(const float* __restrict__ x, const float* __restrict__ rout_W,
                      const float* __restrict__ rout_b, float* __restrict__ out, int N) {
    int i = blockIdx.x * blockDim.x + threadIdx.x;
    if (i >= N) return;
    float a0 = rout_b[0], a1 = rout_b[1];
#pragma unroll
    for (int c = 0; c < 32; ++c) {
        float xv = x[i * 32 + c];
        a0 += xv * rout_W[c * 2 + 0];
        a1 += xv * rout_W[c * 2 + 1];
    }
    out[i * 2 + 0] = a0;
    out[i * 2 + 1] = a1;
}

// ---------------------------------------------------------------------------
static inline int cdiv(long long a, long long b) { return (int)((a + b - 1) / b); }

extern "C" void kernel_launch(void* const* d_in, const int* in_sizes, int n_in,
                              void* d_out, int out_size, void* d_ws, size_t ws_size,
                              hipStream_t stream) {
    const float* own_obs    = (const float*)d_in[0];
    const float* agent_pos  = (const float*)d_in[1];
    const float* target_pos = (const float*)d_in[2];
    const int*   agent_edges = (const int*)d_in[3];
    const int*   t_src      = (const int*)d_in[4];
    const int*   t_dst      = (const int*)d_in[5];
    const int*   gnn_edges  = (const int*)d_in[6];
    const float* edge_attr  = (const float*)d_in[7];
    const float* phiA_W = (const float*)d_in[8];
    const float* phiA_b = (const float*)d_in[9];
    const float* rhoA_W = (const float*)d_in[10];
    const float* rhoA_b = (const float*)d_in[11];
    const float* phiT_W = (const float*)d_in[12];
    const float* phiT_b = (const float*)d_in[13];
    const float* rhoT_W = (const float*)d_in[14];
    const float* rhoT_b = (const float*)d_in[15];
    const float* rin_W  = (const float*)d_in[16];
    const float* rin_b  = (const float*)d_in[17];
    const float* gnn_W  = (const float*)d_in[18];
    const float* gnn_b  = (const float*)d_in[19];
    const float* rout_W = (const float*)d_in[20];
    const float* rout_b = (const float*)d_in[21];

    const int N = in_sizes[0] / 4;             // own_obs is N x 4
    const int E = in_sizes[4];                 // t_src is (E,)
    const int H = in_sizes[9];                 // phiA_b is (H,)
    const int Lnum = in_sizes[19] / 32;        // gnn_b is (L, 32)
    const int KT = in_sizes[18] / (Lnum * 32 * 32);  // TAPS+1
    const int Npad = ((N + 63) / 64) * 64;     // multiple of 64 rows (gemm block)

    float* ws = (float*)d_ws;
    size_t off = 0;
    float* S_A   = ws + off; off += (size_t)2 * N;
    float* degA  = ws + off; off += (size_t)N;
    float* S_T   = ws + off; off += (size_t)2 * N;
    float* degT  = ws + off; off += (size_t)N;
    float* consts = ws + off; off += 64;
    float* xbuf  = ws + off; off += (size_t)Npad * 32;
    float* zbuf  = ws + off; off += (size_t)Npad * 32;
    float* sbufA = ws + off; off += (size_t)Npad * 32;
    float* sbufB = ws + off; off += (size_t)Npad * 32;
    (void)ws_size; (void)n_in; (void)out_size;

    const int TB = 256;

    // accumulators (S_A, degA, S_T, degT are contiguous: 6N floats)
    zero_k<<<cdiv(6LL * N, TB), TB, 0, stream>>>(S_A, 6 * N);
    // collapsed DeepSet constants
    precompute_consts<<<1, 64, 0, stream>>>(phiA_W, phiA_b, rhoA_W,
                                            phiT_W, phiT_b, rhoT_W, consts, H);
    // segment sums (positions + degrees)
    edge_accum<<<cdiv(E, TB), TB, 0, stream>>>(agent_edges, agent_edges + E, agent_pos,
                                               S_A, degA, E);
    edge_accum<<<cdiv(E, TB), TB, 0, stream>>>(t_src, t_dst, target_pos, S_T, degT, E);
    // fused DeepSet-readout + input projection -> x (N x 32)
    node_features<<<cdiv(N, TB), TB, 0, stream>>>(own_obs, agent_pos, S_A, degA, S_T, degT,
                                                  consts, rhoA_b, rhoT_b, rin_W, rin_b,
                                                  xbuf, N);

    // GNN: z = x@W0 + b; for k: s <- seg(w*s[g0], g1); z += s@Wk; x = lrelu(z)
    for (int l = 0; l < Lnum; ++l) {
        gemm32_wmma<<<Npad / 64, 128, 0, stream>>>(
            xbuf, gnn_W + (size_t)(l * KT) * 1024, gnn_b + l * 32, zbuf, 1);
        const float* sprev = xbuf;
        float* snext = sbufA;
        for (int k = 1; k < KT; ++k) {
            zero_k<<<cdiv((long long)N * 32, TB), TB, 0, stream>>>(snext, N * 32);
            propagate<<<cdiv((long long)E * 32, TB), TB, 0, stream>>>(
                sprev, snext, gnn_edges, gnn_edges + E, edge_attr, E);
            gemm32_wmma<<<Npad / 64, 128, 0, stream>>>(
                snext, gnn_W + (size_t)(l * KT + k) * 1024, gnn_b + l * 32, zbuf, 0);
            sprev = snext;
            snext = (snext == sbufA) ? sbufB : sbufA;
        }
        lrelu_k<<<cdiv((long long)N * 32, TB), TB, 0, stream>>>(zbuf, xbuf, N * 32);
    }

    out_k<<<cdiv(N, TB), TB, 0, stream>>>(xbuf, rout_W, rout_b, (float*)d_out, N);
}